// AdditiveAttention_12180527251650
// MI455X (gfx1250) — compile-verified
//
#include <hip/hip_runtime.h>

// AdditiveAttention, MI455X/gfx1250.
// B=4, Q=K=512, D=256, H=128, V=256.  f32 throughout; matrix work on
// V_WMMA_F32_16X16X4_F32; score pass on native V_TANH_F32 (TRANS pipe,
// co-executes with the VALU FMA chain).  Never materializes the
// [B,Q,K,H] tanh tensor (536 MB) nor the scores in HBM.

typedef __attribute__((ext_vector_type(2))) float v2f;
typedef __attribute__((ext_vector_type(8))) float v8f;

#define BB 4
#define QQ 512
#define KKEYS 512
#define DD 256
#define VV 256
#define HH 128
#define MASK_VALUE (-1000000.0f)

#if defined(__gfx1250__) && __has_builtin(__builtin_amdgcn_tanhf)
__device__ __forceinline__ float fast_tanh(float x) { return __builtin_amdgcn_tanhf(x); }
#else
__device__ __forceinline__ float fast_tanh(float x) { return tanhf(x); }
#endif

// ---------------------------------------------------------------------------
// Kernel 1: fused Q/K linear projections with WMMA f32 16x16x4.
//   z==0:  fq[m][h]      = queries_flat[m][:] . wq_w[h][:] + wq_b[h]   ([2048][128])
//   z==1:  fkT[b][h][k]  = keys_flat[m][:]    . wk_w[h][:] + wk_b[h]   (transposed)
// One wave per 16x16 output tile, 64 WMMA steps over D=256.
// ---------------------------------------------------------------------------
__global__ __launch_bounds__(32) void proj_wmma_kernel(
    const float* __restrict__ queries, const float* __restrict__ keys,
    const float* __restrict__ wq_w, const float* __restrict__ wq_b,
    const float* __restrict__ wk_w, const float* __restrict__ wk_b,
    float* __restrict__ fq, float* __restrict__ fkT)
{
  const bool isK = (blockIdx.z != 0);
  const float* __restrict__ X    = isK ? keys : queries;
  const float* __restrict__ W    = isK ? wk_w : wq_w;   // [H][D] row-major
  const float* __restrict__ bias = isK ? wk_b : wq_b;

  const int m0 = blockIdx.x * 16;      // row tile over B*Q (or B*K) = 2048
  const int h0 = blockIdx.y * 16;      // col tile over H = 128
  const int lane = threadIdx.x;
  const int r  = lane & 15;            // A row in tile == B/C column in tile
  const int kh = (lane >> 4) * 2;      // K sub-offset for this half-wave

  const float* Xp = X + (size_t)(m0 + r) * DD + kh;       // A: row-major rows
  const float* Wp = W + (size_t)(h0 + r) * DD + kh;       // B[k][n] = W[h0+n][k]

  v8f acc = {};
  #pragma unroll 4
  for (int k = 0; k < DD; k += 4) {
    v2f a = *(const v2f*)(Xp + k);     // A(m, k+kh), A(m, k+kh+1)
    v2f b = *(const v2f*)(Wp + k);     // B(k+kh, n), B(k+kh+1, n)
    acc = __builtin_amdgcn_wmma_f32_16x16x4_f32(false, a, false, b,
                                                (short)0, acc, false, false);
  }

  const float bv  = bias[h0 + r];
  const int   mhi = (lane >> 4) * 8;   // C layout: lanes 16-31 hold M=8..15
  if (!isK) {
    #pragma unroll
    for (int v = 0; v < 8; ++v)
      fq[(size_t)(m0 + mhi + v) * HH + h0 + r] = acc[v] + bv;
  } else {
    #pragma unroll
    for (int v = 0; v < 8; ++v) {
      const int row = m0 + mhi + v;    // global key row in [0, B*K)
      const int b_  = row >> 9;        // / 512
      const int kk  = row & 511;       // consecutive over v -> b128 stores
      fkT[((size_t)b_ * HH + h0 + r) * KKEYS + kk] = acc[v] + bv;
    }
  }
}

// ---------------------------------------------------------------------------
// Kernel 2: fused scores (tanh) + masked softmax + attn@values.
// Block = 512 threads = one (batch, 16-query tile); thread t owns key k=t.
// ---------------------------------------------------------------------------
__global__ __launch_bounds__(512) void attn_fused_kernel(
    const float* __restrict__ fq, const float* __restrict__ fkT,
    const float* __restrict__ values, const int* __restrict__ valid_lens,
    const float* __restrict__ wv_w, const float* __restrict__ wv_b,
    float* __restrict__ out)
{
  __shared__ float fq_s[16][HH];
  __shared__ float wv_s[HH];
  __shared__ float attn_s[16][KKEYS + 4];   // stride 516: conflict-free A-frags
  __shared__ float red_s[16][16];
  __shared__ float mx_s[16];
  __shared__ float inv_s[16];

  const int b  = blockIdx.x >> 5;           // 32 q-tiles per batch
  const int q0 = (blockIdx.x & 31) * 16;
  const int t    = threadIdx.x;             // key index
  const int wave = t >> 5;
  const int lane = t & 31;

  for (int i = t; i < 16 * HH; i += 512) {
    const int q = i >> 7, h = i & (HH - 1);
    fq_s[q][h] = fq[(size_t)(b * QQ + q0 + q) * HH + h];
  }
  if (t < HH) wv_s[t] = wv_w[t];
  __syncthreads();

  // ---- scores[q][t] = sum_h wv[h]*tanh(fq[q][h] + fk[t][h]) ----
  // Inner body per h: 16 x (v_add_f32 -> v_tanh_f32 -> v_fmac_f32); the
  // TRANS tanh co-executes with the VALU adds/FMAs of neighboring q's.
  float s[16];
  #pragma unroll
  for (int q = 0; q < 16; ++q) s[q] = 0.f;
  const float* fkcol = fkT + (size_t)b * HH * KKEYS + t;   // coalesced over t
  for (int h = 0; h < HH; ++h) {
    const float fkv = fkcol[(size_t)h * KKEYS];
    const float wvh = wv_s[h];
    #pragma unroll
    for (int q = 0; q < 16; ++q)
      s[q] = fmaf(wvh, fast_tanh(fq_s[q][h] + fkv), s[q]);
  }
  const float sb    = wv_b[0];
  const bool  valid = t < valid_lens[b];
  #pragma unroll
  for (int q = 0; q < 16; ++q) s[q] = valid ? (s[q] + sb) : MASK_VALUE;

  // ---- masked softmax over k: wave32 shuffle tree + LDS across 16 waves ----
  #pragma unroll
  for (int q = 0; q < 16; ++q) {
    float v = s[q];
    #pragma unroll
    for (int off = 16; off > 0; off >>= 1) v = fmaxf(v, __shfl_xor(v, off, 32));
    if (lane == 0) red_s[q][wave] = v;
  }
  __syncthreads();
  if (t < 16) {
    float v = red_s[t][0];
    #pragma unroll
    for (int w = 1; w < 16; ++w) v = fmaxf(v, red_s[t][w]);
    mx_s[t] = v;
  }
  __syncthreads();
  float p[16];
  #pragma unroll
  for (int q = 0; q < 16; ++q) p[q] = __expf(s[q] - mx_s[q]);
  __syncthreads();                         // red_s reuse
  #pragma unroll
  for (int q = 0; q < 16; ++q) {
    float v = p[q];
    #pragma unroll
    for (int off = 16; off > 0; off >>= 1) v += __shfl_xor(v, off, 32);
    if (lane == 0) red_s[q][wave] = v;
  }
  __syncthreads();
  if (t < 16) {
    float v = 0.f;
    #pragma unroll
    for (int w = 0; w < 16; ++w) v += red_s[t][w];
    inv_s[t] = 1.0f / v;
  }
  __syncthreads();
  #pragma unroll
  for (int q = 0; q < 16; ++q) attn_s[q][t] = p[q] * inv_s[q];
  __syncthreads();

  // ---- AV: out[b, q0:q0+16, :] = attn(16x512) @ values[b](512x256) ----
  // 16 waves, wave w owns V-columns [16w, 16w+16); 128 WMMA k-steps.
  const int n0  = wave * 16;
  const int col = lane & 15;
  const int kh  = (lane >> 4) * 2;
  const float* Vp = values + (size_t)b * KKEYS * VV + n0 + col;
  v8f acc = {};
  for (int k = 0; k < KKEYS; k += 4) {
    v2f a = *(const v2f*)(&attn_s[col][k + kh]);           // ds_load_b64
    v2f bf;
    bf[0] = Vp[(size_t)(k + kh) * VV];
    bf[1] = Vp[(size_t)(k + kh + 1) * VV];
    acc = __builtin_amdgcn_wmma_f32_16x16x4_f32(false, a, false, bf,
                                                (short)0, acc, false, false);
  }
  const int mhi = (lane >> 4) * 8;
  float* op = out + (size_t)(b * QQ + q0 + mhi) * VV + n0 + col;
  #pragma unroll
  for (int v = 0; v < 8; ++v)
    op[(size_t)v * VV] = acc[v];
}

// ---------------------------------------------------------------------------
extern "C" void kernel_launch(void* const* d_in, const int* in_sizes, int n_in,
                              void* d_out, int out_size, void* d_ws, size_t ws_size,
                              hipStream_t stream) {
  (void)in_sizes; (void)n_in; (void)out_size; (void)ws_size;
  const float* queries    = (const float*)d_in[0];  // [4,512,256]
  const float* keys       = (const float*)d_in[1];  // [4,512,256]
  const float* values     = (const float*)d_in[2];  // [4,512,256]
  const int*   valid_lens = (const int*)  d_in[3];  // [4]
  const float* wq_w       = (const float*)d_in[4];  // [128,256]
  const float* wq_b       = (const float*)d_in[5];  // [128]
  const float* wk_w       = (const float*)d_in[6];  // [128,256]
  const float* wk_b       = (const float*)d_in[7];  // [128]
  const float* wv_w       = (const float*)d_in[8];  // [1,128]
  const float* wv_b       = (const float*)d_in[9];  // [1]
  float* out = (float*)d_out;                       // [4,512,256]

  float* fq  = (float*)d_ws;                        // [2048][128]  (1 MB)
  float* fkT = fq + (size_t)BB * KKEYS * HH;        // [4][128][512] (1 MB)

  dim3 g1((BB * QQ) / 16, HH / 16, 2);              // 128 x 8 x 2 tiles
  proj_wmma_kernel<<<g1, 32, 0, stream>>>(queries, keys, wq_w, wq_b,
                                          wk_w, wk_b, fq, fkT);

  attn_fused_kernel<<<BB * (QQ / 16), 512, 0, stream>>>(
      fq, fkT, values, valid_lens, wv_w, wv_b, out);
}